// Solver_31464930410650
// MI455X (gfx1250) — compile-verified
//
#include <hip/hip_runtime.h>
#include <math.h>

// ---------------- problem constants ----------------
constexpr int   NXg  = 32768;
constexpr int   NVg  = 16384;
constexpr int   OST  = NXg + 64 + NVg;   // output row stride (49216)
constexpr float LXf  = 1.0f;
constexpr float VMAXf= 8.0f;
constexpr float DXf  = LXf / NXg;
constexpr float DVf  = 2.0f * VMAXf / NVg;
constexpr float DTf  = 1e-3f;
constexpr float AEf  = 1.0f,  AIf = 1836.0f;
constexpr float ZEf  = -1.0f, ZIf = 1.0f;
constexpr float NUEf = 1.0f,  NUIf = 0.04f;
constexpr float OCTf = 1.0f;             // omega_c_tau
constexpr float OP2f = 1.0f;             // omega_p_tau^2
constexpr int   GRAM_CHUNK = 2048;
constexpr int   LP = 66;                 // LDS row pitch (even -> b64-aligned frags)

typedef __attribute__((ext_vector_type(2))) float v2f;
typedef __attribute__((ext_vector_type(8))) float v8f;

// ---------------- device helpers ----------------
__device__ __forceinline__ float vs_of(int v) {
  return -VMAXf + (v + 0.5f) * DVf;
}
__device__ __forceinline__ float maxw(float A, int v) {
  float x = vs_of(v);
  return sqrtf(A * 0.15915494309189535f) * expf(-0.5f * A * x * x);
}
__device__ __forceinline__ v8f wmma4(v2f a, v2f b, v8f c) {
  return __builtin_amdgcn_wmma_f32_16x16x4_f32(false, a, false, b, (short)0, c,
                                               false, false);
}

// =======================================================================
// WMMA GEMM: C(64xN) = alpha*op(A)(64x64) @ Beff(64xN) + beta*C
//  MODE 0: Beff[k,n] = B[k,n]
//  MODE 1: Beff[k,n] = B[k,n+S1] - B[k,n+S0]   (upwind flux difference)
//  MODE 2: Beff[k,n] = B[k,n] * cs[n]          (E-field / v± column scale)
// block = 256 threads (8 waves); each wave one 16x16 tile; block tile 64x32.
// N, MODE, S0, S1 compile-time -> immediate-offset loads; boundary checks
// exist only in the (uniform) edge blocks of MODE 1.
// =======================================================================
template <int MODE, int S0, int S1, int N>
__global__ __launch_bounds__(256)
void gemm64_wmma(const float* __restrict__ A, int transA,
                 const float* __restrict__ B, float* __restrict__ C,
                 float alpha, float beta, const float* __restrict__ cs) {
  __shared__ float As[64 * LP];
  for (int idx = threadIdx.x; idx < 64 * 64; idx += 256) {
    int i = idx >> 6, k = idx & 63;
    As[i * LP + k] = transA ? A[k * 64 + i] : A[idx];
  }
  __syncthreads();
  const int wave = threadIdx.x >> 5;
  const int lane = threadIdx.x & 31;
  const int half = lane >> 4;
  const int l    = lane & 15;
  const int m0   = (wave >> 1) * 16;
  const int col  = blockIdx.x * 32 + (wave & 1) * 16 + l;
  const int ko   = half ? 2 : 0;          // k-offset of this half's pair
  const float* Arow = As + (m0 + l) * LP;
  const bool edge =
      (MODE == 1) && (blockIdx.x == 0 || blockIdx.x == (N / 32 - 1));
  v8f acc = {};
  if (MODE == 1 && edge) {
#pragma unroll
    for (int k0 = 0; k0 < 64; k0 += 4) {
      v2f a = *(const v2f*)(Arow + k0 + ko);
      v2f b;
      int c0 = col + S0, c1 = col + S1;
      float x0 = (c0 >= 0 && c0 < N) ? B[(k0 + ko) * N + c0] : 0.f;
      float x1 = (c1 >= 0 && c1 < N) ? B[(k0 + ko) * N + c1] : 0.f;
      float y0 = (c0 >= 0 && c0 < N) ? B[(k0 + ko + 1) * N + c0] : 0.f;
      float y1 = (c1 >= 0 && c1 < N) ? B[(k0 + ko + 1) * N + c1] : 0.f;
      b.x = x1 - x0;
      b.y = y1 - y0;
      acc = wmma4(a, b, acc);
    }
  } else {
    const float csv = (MODE == 2) ? cs[col] : 1.f;
    const float* Bc = B + col;
#pragma unroll
    for (int k0 = 0; k0 < 64; k0 += 4) {
      v2f a = *(const v2f*)(Arow + k0 + ko);
      v2f b;
      if (MODE == 0) {
        b.x = Bc[(k0 + ko) * N];
        b.y = Bc[(k0 + ko + 1) * N];
      } else if (MODE == 1) {
        b.x = Bc[(k0 + ko) * N + S1] - Bc[(k0 + ko) * N + S0];
        b.y = Bc[(k0 + ko + 1) * N + S1] - Bc[(k0 + ko + 1) * N + S0];
      } else {
        b.x = Bc[(k0 + ko) * N] * csv;
        b.y = Bc[(k0 + ko + 1) * N] * csv;
      }
      acc = wmma4(a, b, acc);
    }
  }
#pragma unroll
  for (int j = 0; j < 8; ++j) {
    int row = m0 + j + (half ? 8 : 0);
    float r = alpha * acc[j];
    if (beta != 0.f) r += beta * C[(size_t)row * N + col];
    C[(size_t)row * N + col] = r;
  }
}

// =======================================================================
// WMMA Gram: part[block] = A(64 x chunk) @ Beff(64 x chunk)^T  (64x64 partial)
// 64-column panels staged to LDS (float2-vectorized where alignment allows,
// mode transform applied during staging, rows padded to LP floats so the
// fragment reads are single aligned ds_load_b64). 8 waves x 2 tiles cover the
// 64x64 output; 32 WMMA per panel. Deterministic fixed-order reduction.
// =======================================================================
template <int MODE, int S0, int S1, int N>
__global__ __launch_bounds__(256)
void gram64_wmma(const float* __restrict__ A, const float* __restrict__ B,
                 float* __restrict__ part, const float* __restrict__ cs) {
  __shared__ float As[64 * LP];
  __shared__ float Bs[64 * LP];
  const int wave = threadIdx.x >> 5;
  const int lane = threadIdx.x & 31;
  const int half = lane >> 4;
  const int l    = lane & 15;
  const int ko   = half ? 2 : 0;
  const int i00 = (wave >> 2) * 16,       j00 = (wave & 3) * 16;
  const int i01 = ((wave + 8) >> 2) * 16, j01 = ((wave + 8) & 3) * 16;
  const float* Ar0 = As + (i00 + l) * LP;
  const float* Ar1 = As + (i01 + l) * LP;
  const float* Br0 = Bs + (j00 + l) * LP;
  const float* Br1 = Bs + (j01 + l) * LP;
  const int nb0 = blockIdx.x * GRAM_CHUNK;
  const bool edge =
      (MODE == 1) && (blockIdx.x == 0 || blockIdx.x == (N / GRAM_CHUNK - 1));
  v8f acc0 = {}, acc1 = {};
  for (int pn = 0; pn < GRAM_CHUNK; pn += 64) {
    const int nb = nb0 + pn;
    __syncthreads();
    // stage panel: each thread handles 2 consecutive columns (8B aligned)
    for (int idx = threadIdx.x; idx < 64 * 32; idx += 256) {
      int r = idx >> 5, c2 = (idx & 31) * 2;
      int gc = nb + c2;
      *(v2f*)&As[r * LP + c2] = *(const v2f*)&A[(size_t)r * N + gc];
      if (MODE == 0) {
        *(v2f*)&Bs[r * LP + c2] = *(const v2f*)&B[(size_t)r * N + gc];
      } else if (MODE == 2) {
        v2f bv = *(const v2f*)&B[(size_t)r * N + gc];
        v2f cv = *(const v2f*)&cs[gc];
        bv.x *= cv.x;
        bv.y *= cv.y;
        *(v2f*)&Bs[r * LP + c2] = bv;
      } else {  // MODE 1: shifted difference (odd offsets -> scalar)
#pragma unroll
        for (int u = 0; u < 2; ++u) {
          int gcc = gc + u;
          float va, vb;
          if (!edge) {
            va = B[(size_t)r * N + gcc + S0];
            vb = B[(size_t)r * N + gcc + S1];
          } else {
            int c0 = gcc + S0, c1 = gcc + S1;
            va = (c0 >= 0 && c0 < N) ? B[(size_t)r * N + c0] : 0.f;
            vb = (c1 >= 0 && c1 < N) ? B[(size_t)r * N + c1] : 0.f;
          }
          Bs[r * LP + c2 + u] = vb - va;
        }
      }
    }
    // prefetch next panel (one cacheline per thread) -> global_prefetch_b8
    if (pn + 64 < GRAM_CHUNK) {
      int r = threadIdx.x >> 2, c = (threadIdx.x & 3) * 16;
      __builtin_prefetch(&A[(size_t)r * N + nb + 64 + c], 0, 1);
      __builtin_prefetch(&B[(size_t)r * N + nb + 64 + c], 0, 1);
    }
    __syncthreads();
#pragma unroll
    for (int k0 = 0; k0 < 64; k0 += 4) {
      v2f a = *(const v2f*)(Ar0 + k0 + ko);
      v2f b = *(const v2f*)(Br0 + k0 + ko);
      acc0 = wmma4(a, b, acc0);
      a = *(const v2f*)(Ar1 + k0 + ko);
      b = *(const v2f*)(Br1 + k0 + ko);
      acc1 = wmma4(a, b, acc1);
    }
  }
  float* P = part + (size_t)blockIdx.x * 4096;
#pragma unroll
  for (int j = 0; j < 8; ++j) {
    P[(i00 + j + (half ? 8 : 0)) * 64 + (j00 + l)] = acc0[j];
    P[(i01 + j + (half ? 8 : 0)) * 64 + (j01 + l)] = acc1[j];
  }
}

__global__ void k_gram_reduce(const float* __restrict__ part, int nparts,
                              float alpha, float* __restrict__ G) {
  int idx = blockIdx.x * 256 + threadIdx.x;  // 16 blocks x 256 = 4096
  float s = 0.f;
  for (int p = 0; p < nparts; ++p) s += part[(size_t)p * 4096 + idx];
  G[idx] = alpha * s;
}

// ---------------- small dense helpers ----------------
__global__ void k_mm64(const float* __restrict__ A, int tA,
                       const float* __restrict__ B, int tB,
                       float* __restrict__ C, float alpha, float beta) {
  int idx = blockIdx.x * 256 + threadIdx.x;
  int i = idx >> 6, j = idx & 63;
  float s = 0.f;
  for (int k = 0; k < 64; ++k) {
    float a = tA ? A[k * 64 + i] : A[i * 64 + k];
    float b = tB ? B[j * 64 + k] : B[k * 64 + j];
    s += a * b;
  }
  float o = (beta != 0.f) ? beta * C[idx] : 0.f;
  C[idx] = o + alpha * s;
}

__global__ void k_matvec64(const float* __restrict__ M,
                           const float* __restrict__ v, float* __restrict__ out,
                           float scale) {
  int i = threadIdx.x;
  float s = 0.f;
  for (int j = 0; j < 64; ++j) s += M[i * 64 + j] * v[j];
  out[i] = s * scale;
}

__global__ void k_fluxt1(const float* __restrict__ X,
                         const float* __restrict__ S, float* __restrict__ t1) {
  int j = threadIdx.x;
  float s = 0.f;
  for (int i = 0; i < 64; ++i)
    s += X[(size_t)i * NXg + (NXg - 1)] * S[i * 64 + j];
  t1[j] = s;
}

__global__ void k_dot64(const float* __restrict__ a,
                        const float* __restrict__ b, float* __restrict__ out) {
  __shared__ float sh[64];
  int t = threadIdx.x;
  sh[t] = a[t] * b[t];
  __syncthreads();
  for (int off = 32; off; off >>= 1) {
    if (t < off) sh[t] += sh[t + off];
    __syncthreads();
  }
  if (t == 0) out[0] = sh[0];
}

// out[i] = scale * sum_n A[i,n]*w(n); wmode: 0=array, 1=maxwellian(Am), 2=v+, 3=ones
__global__ void k_rowdot(const float* __restrict__ A, int N,
                         const float* __restrict__ w, int wmode, float Am,
                         float scale, float* __restrict__ out) {
  __shared__ float sh[256];
  int i = blockIdx.x;
  float s = 0.f;
  for (int n = threadIdx.x; n < N; n += 256) {
    float wv;
    if (wmode == 0)      wv = w[n];
    else if (wmode == 1) wv = maxw(Am, n);
    else if (wmode == 2) wv = fmaxf(vs_of(n), 0.f);
    else                 wv = 1.f;
    s += A[(size_t)i * N + n] * wv;
  }
  sh[threadIdx.x] = s;
  __syncthreads();
  for (int off = 128; off; off >>= 1) {
    if (threadIdx.x < off) sh[threadIdx.x] += sh[threadIdx.x + off];
    __syncthreads();
  }
  if (threadIdx.x == 0) out[i] = sh[0] * scale;
}

// out[n] = beta*out[n] + alpha * sum_i A[i,n]*w[i]
__global__ void k_coldot(const float* __restrict__ A, int N,
                         const float* __restrict__ w, float alpha, float beta,
                         float* __restrict__ out) {
  int n = blockIdx.x * 256 + threadIdx.x;
  if (n >= N) return;
  float s = 0.f;
  for (int i = 0; i < 64; ++i) s += A[(size_t)i * N + n] * w[i];
  float o = (beta != 0.f) ? beta * out[n] : 0.f;
  out[n] = o + alpha * s;
}

// ---------------- Poisson: E = cumsum(rho)*dx*wp^2, then subtract mean ----
__global__ __launch_bounds__(1024)
void k_scan_E(const float* __restrict__ rho, float* __restrict__ E,
              float* __restrict__ mean) {
  __shared__ float sh[1024];
  __shared__ float carry;
  if (threadIdx.x == 0) carry = 0.f;
  __syncthreads();
  float sumE = 0.f;
  for (int base = 0; base < NXg; base += 1024) {
    sh[threadIdx.x] = rho[base + threadIdx.x];
    __syncthreads();
    for (int off = 1; off < 1024; off <<= 1) {
      float t = (threadIdx.x >= off) ? sh[threadIdx.x - off] : 0.f;
      __syncthreads();
      sh[threadIdx.x] += t;
      __syncthreads();
    }
    float c = carry;
    float e = (c + sh[threadIdx.x]) * (DXf * OP2f);
    E[base + threadIdx.x] = e;
    sumE += e;
    __syncthreads();
    if (threadIdx.x == 1023) carry = c + sh[1023];
    __syncthreads();
  }
  sh[threadIdx.x] = sumE;
  __syncthreads();
  for (int off = 512; off; off >>= 1) {
    if (threadIdx.x < off) sh[threadIdx.x] += sh[threadIdx.x + off];
    __syncthreads();
  }
  if (threadIdx.x == 0) mean[0] = sh[0] / (float)NXg;
}

__global__ void k_submean(float* __restrict__ E, const float* __restrict__ m) {
  int x = blockIdx.x * 256 + threadIdx.x;
  E[x] -= m[0];
}

__global__ void k_epem(const float* __restrict__ E, float* __restrict__ ep,
                       float* __restrict__ em, float fac) {
  int x = blockIdx.x * 256 + threadIdx.x;
  float v = fac * E[x];
  ep[x] = fmaxf(v, 0.f);
  em[x] = fminf(v, 0.f);
}

// ---------------- Cholesky-QR pieces ----------------
__global__ __launch_bounds__(64)
void k_chol64(float* __restrict__ G) {
  __shared__ float s[64 * 64];
  int t = threadIdx.x;
  for (int idx = t; idx < 4096; idx += 64) s[idx] = G[idx];
  __syncthreads();
  for (int k = 0; k < 64; ++k) {
    if (t == k) s[k * 64 + k] = sqrtf(s[k * 64 + k]);
    __syncthreads();
    if (t > k) s[t * 64 + k] /= s[k * 64 + k];
    __syncthreads();
    for (int j = k + 1; j <= t; ++j) s[t * 64 + j] -= s[t * 64 + k] * s[j * 64 + k];
    __syncthreads();
  }
  for (int idx = t; idx < 4096; idx += 64) {
    int i = idx >> 6, j = idx & 63;
    G[idx] = (j <= i) ? s[idx] : 0.f;
  }
}

// X = scale * L^{-1} K : one column per lane, forward substitution.
__global__ __launch_bounds__(256)
void k_trsm(const float* __restrict__ K, float* __restrict__ X, int N,
            const float* __restrict__ L, float scale) {
  __shared__ float Ls[64 * 64];
  for (int idx = threadIdx.x; idx < 4096; idx += 256) Ls[idx] = L[idx];
  __syncthreads();
  int x = blockIdx.x * 256 + threadIdx.x;
  if (x >= N) return;
  float y[64];
#pragma unroll
  for (int i = 0; i < 64; ++i) {
    float s = K[(size_t)i * N + x];
    for (int j = 0; j < i; ++j) s -= Ls[i * 64 + j] * y[j];
    y[i] = s / Ls[i * 64 + i];
  }
#pragma unroll
  for (int i = 0; i < 64; ++i) X[(size_t)i * N + x] = y[i] * scale;
}

__global__ void k_transcale(const float* __restrict__ L, float* __restrict__ out,
                            int trans, float s) {
  int idx = blockIdx.x * 256 + threadIdx.x;
  int i = idx >> 6, j = idx & 63;
  out[idx] = (trans ? L[j * 64 + i] : L[idx]) * s;
}

// ---------------- init / elementwise ----------------
__global__ void k_init_vpm(float* __restrict__ vp, float* __restrict__ vm) {
  int v = blockIdx.x * 256 + threadIdx.x;
  float s = vs_of(v);
  vp[v] = fmaxf(s, 0.f);
  vm[v] = fminf(s, 0.f);
}

__global__ void k_gauss(float* __restrict__ g) {
  int x = blockIdx.x * 256 + threadIdx.x;
  float xs = (x + 0.5f) * DXf - 0.5f * LXf;
  float r = xs / (0.1f * LXf);
  g[x] = expf(-0.5f * r * r);
}

__global__ __launch_bounds__(1024)
void k_reduce_sum(const float* __restrict__ a, int N, float* __restrict__ out) {
  __shared__ float sh[1024];
  float s = 0.f;
  for (int n = threadIdx.x; n < N; n += 1024) s += a[n];
  sh[threadIdx.x] = s;
  __syncthreads();
  for (int off = 512; off; off >>= 1) {
    if (threadIdx.x < off) sh[threadIdx.x] += sh[threadIdx.x + off];
    __syncthreads();
  }
  if (threadIdx.x == 0) out[0] = sh[0];
}

__global__ void k_scale_gshape(float* __restrict__ g, const float* __restrict__ s) {
  int x = blockIdx.x * 256 + threadIdx.x;
  g[x] = g[x] / (s[0] * DXf);
}

__global__ void k_makew(const float* __restrict__ n, const float* __restrict__ g,
                        const float* __restrict__ flx, float nu0,
                        float* __restrict__ w) {
  int x = blockIdx.x * 256 + threadIdx.x;
  w[x] = n[x] * nu0 + flx[0] * g[x];
}

// K += dt*(RHS + (n*nu0 + flux*g)*VM[r] - K*nu0)
__global__ void k_finishK(float* __restrict__ K, const float* __restrict__ RHS,
                          const float* __restrict__ nb, const float* __restrict__ g,
                          const float* __restrict__ VM, const float* __restrict__ flx,
                          float nu0) {
  int x = blockIdx.x * 256 + threadIdx.x;
  int r = blockIdx.y;
  size_t idx = (size_t)r * NXg + x;
  float coll = (nb[x] * nu0 + flx[0] * g[x]) * VM[r] - K[idx] * nu0;
  K[idx] += DTf * (RHS[idx] + coll);
}

// S += dt*(T1 - Xng[i]*VM[j])
__global__ void k_finishS(float* __restrict__ S, const float* __restrict__ T1,
                          const float* __restrict__ Xng, const float* __restrict__ VM) {
  int idx = blockIdx.x * 256 + threadIdx.x;
  int i = idx >> 6, j = idx & 63;
  S[idx] += DTf * (T1[idx] - Xng[i] * VM[j]);
}

// L += dt*(RHS + Xng[i]*Maxwellian(v))
__global__ void k_finishL(float* __restrict__ L, const float* __restrict__ RHS,
                          const float* __restrict__ Xng, float Am) {
  int v = blockIdx.x * 256 + threadIdx.x;
  int r = blockIdx.y;
  size_t idx = (size_t)r * NVg + v;
  L[idx] += DTf * (RHS[idx] + Xng[r] * maxw(Am, v));
}

__global__ void k_copyout(const float* __restrict__ src, int ncols,
                          float* __restrict__ dst, int coloff) {
  int c = blockIdx.x * 256 + threadIdx.x;
  int r = blockIdx.y;
  if (c < ncols) dst[(size_t)r * OST + coloff + c] = src[(size_t)r * ncols + c];
}

// ---------------- host-side templated launch helpers ----------------
template <int MODE, int S0, int S1, int N>
static inline void gemmT(hipStream_t st, const float* A, int tA, const float* B,
                         float* C, float al, float be, const float* cs) {
  gemm64_wmma<MODE, S0, S1, N>
      <<<dim3(N / 32), dim3(256), 0, st>>>(A, tA, B, C, al, be, cs);
}
template <int MODE, int S0, int S1, int N>
static inline void gramT(hipStream_t st, const float* A, const float* B,
                         float* GP, float al, float* G, const float* cs) {
  gram64_wmma<MODE, S0, S1, N>
      <<<dim3(N / GRAM_CHUNK), dim3(256), 0, st>>>(A, B, GP, cs);
  k_gram_reduce<<<dim3(16), dim3(256), 0, st>>>(GP, N / GRAM_CHUNK, al, G);
}

// ---------------- orchestration ----------------
extern "C" void kernel_launch(void* const* d_in, const int* in_sizes, int n_in,
                              void* d_out, int out_size, void* d_ws,
                              size_t ws_size, hipStream_t stream) {
  (void)in_sizes; (void)n_in; (void)out_size; (void)ws_size;
  const float* Xe = (const float*)d_in[0];
  const float* Se = (const float*)d_in[1];
  const float* Ve = (const float*)d_in[2];
  const float* Xi = (const float*)d_in[3];
  const float* Si = (const float*)d_in[4];
  const float* Vi = (const float*)d_in[5];
  float* out = (float*)d_out;

  float* w = (float*)d_ws;
  size_t o = 0;
  auto W = [&](size_t n) { float* p = w + o; o += n; return p; };
  float* Ke   = W((size_t)64 * NXg);
  float* Ki   = W((size_t)64 * NXg);
  float* BUFX = W((size_t)64 * NXg);  // RHS (K step) / Kf (S,L steps)
  float* Le   = W((size_t)64 * NVg);
  float* Li   = W((size_t)64 * NVg);
  float* RHSL = W((size_t)64 * NVg);
  float* Vt   = W((size_t)64 * NVg);
  float* Eb   = W(NXg);
  float* rho  = W(NXg);
  float* gsh  = W(NXg);
  float* nbuf = W(NXg);
  float* wvec = W(NXg);
  float* epe  = W(NXg); float* eme = W(NXg);
  float* epi  = W(NXg); float* emi = W(NXg);
  float* vpl  = W(NVg); float* vmn = W(NVg);
  float* GP   = W((size_t)16 * 4096);   // gram partials
  float* G    = W(4096);
  float* vpm_e = W(4096); float* vmm_e = W(4096);
  float* Vl_e  = W(4096); float* Vr_e  = W(4096);
  float* vpm_i = W(4096); float* vmm_i = W(4096);
  float* Vl_i  = W(4096); float* Vr_i  = W(4096);
  float* Epm = W(4096); float* Emm = W(4096);
  float* Klm = W(4096); float* Krm = W(4096);
  float* T1  = W(4096); float* T2  = W(4096);
  float* Xnm = W(4096); float* Sq  = W(4096);
  float* Se_c = W(4096); float* Si_c = W(4096);
  float* sve = W(64); float* svi = W(64); float* wvi = W(64);
  float* VMe = W(64); float* VMi = W(64);
  float* t64 = W(64); float* t1v = W(64);
  float* Xng = W(64); float* lse = W(64); float* lsi = W(64);
  float* flx = W(64); float* mean = W(64); float* gsum = W(64);

  const dim3 B256(256);
  auto mm = [&](const float* A, int tA, const float* Bm, int tB, float* C,
                float al, float be) {
    k_mm64<<<dim3(16), B256, 0, stream>>>(A, tA, Bm, tB, C, al, be);
  };
  const float sDX = sqrtf(DXf), isDX = 1.0f / sqrtf(DXf);
  const float sDV = sqrtf(DVf), isDV = 1.0f / sqrtf(DVf);

  // ---------- init ----------
  k_init_vpm<<<dim3(NVg / 256), B256, 0, stream>>>(vpl, vmn);
  k_gauss<<<dim3(NXg / 256), B256, 0, stream>>>(gsh);
  k_reduce_sum<<<dim3(1), dim3(1024), 0, stream>>>(gsh, NXg, gsum);
  k_scale_gshape<<<dim3(NXg / 256), B256, 0, stream>>>(gsh, gsum);
  k_rowdot<<<dim3(64), B256, 0, stream>>>(Ve, NVg, nullptr, 3, 0.f, 1.f, sve);
  k_rowdot<<<dim3(64), B256, 0, stream>>>(Vi, NVg, nullptr, 3, 0.f, 1.f, svi);
  k_rowdot<<<dim3(64), B256, 0, stream>>>(Vi, NVg, nullptr, 2, 0.f, DVf, wvi);
  k_rowdot<<<dim3(64), B256, 0, stream>>>(Ve, NVg, nullptr, 1, AEf, DVf, VMe);
  k_rowdot<<<dim3(64), B256, 0, stream>>>(Vi, NVg, nullptr, 1, AIf, DVf, VMi);
  gramT<2, 0, 0, NVg>(stream, Ve, Ve, GP, DVf, vpm_e, vpl);
  gramT<2, 0, 0, NVg>(stream, Ve, Ve, GP, DVf, vmm_e, vmn);
  gramT<1, -1, 0, NVg>(stream, Ve, Ve, GP, 1.f, Vl_e, nullptr);
  gramT<1, 0, 1, NVg>(stream, Ve, Ve, GP, 1.f, Vr_e, nullptr);
  gramT<2, 0, 0, NVg>(stream, Vi, Vi, GP, DVf, vpm_i, vpl);
  gramT<2, 0, 0, NVg>(stream, Vi, Vi, GP, DVf, vmm_i, vmn);
  gramT<1, -1, 0, NVg>(stream, Vi, Vi, GP, 1.f, Vl_i, nullptr);
  gramT<1, 0, 1, NVg>(stream, Vi, Vi, GP, 1.f, Vr_i, nullptr);

  // ---------- K step ----------
  k_fluxt1<<<dim3(1), dim3(64), 0, stream>>>(Xi, Si, t1v);
  k_dot64<<<dim3(1), dim3(64), 0, stream>>>(t1v, wvi, flx);
  gemmT<0, 0, 0, NXg>(stream, Se, 1, Xe, Ke, 1.f, 0.f, nullptr);
  gemmT<0, 0, 0, NXg>(stream, Si, 1, Xi, Ki, 1.f, 0.f, nullptr);
  k_coldot<<<dim3(NXg / 256), B256, 0, stream>>>(Ke, NXg, sve, ZEf * DVf, 0.f, rho);
  k_coldot<<<dim3(NXg / 256), B256, 0, stream>>>(Ki, NXg, svi, ZIf * DVf, 1.f, rho);
  k_scan_E<<<dim3(1), dim3(1024), 0, stream>>>(rho, Eb, mean);
  k_submean<<<dim3(NXg / 256), B256, 0, stream>>>(Eb, mean);
  k_epem<<<dim3(NXg / 256), B256, 0, stream>>>(Eb, epe, eme, OCTf * ZEf / AEf);
  k_epem<<<dim3(NXg / 256), B256, 0, stream>>>(Eb, epi, emi, OCTf * ZIf / AIf);
  auto kstep = [&](float* K, const float* vpm, const float* vmm, const float* Vl,
                   const float* Vr, const float* sv, const float* VM,
                   const float* ep, const float* em, float nu0) {
    gemmT<1, -1, 0, NXg>(stream, vpm, 0, K, BUFX, -1.f / DXf, 0.f, nullptr);
    gemmT<1, 0, 1, NXg>(stream, vmm, 0, K, BUFX, -1.f / DXf, 1.f, nullptr);
    gemmT<2, 0, 0, NXg>(stream, Vl, 0, K, BUFX, -1.f, 1.f, ep);
    gemmT<2, 0, 0, NXg>(stream, Vr, 0, K, BUFX, -1.f, 1.f, em);
    k_coldot<<<dim3(NXg / 256), B256, 0, stream>>>(K, NXg, sv, DVf, 0.f, nbuf);
    k_finishK<<<dim3(NXg / 256, 64), B256, 0, stream>>>(K, BUFX, nbuf, gsh, VM,
                                                        flx, nu0);
    // Cholesky-QR: K -> X (in place), S = L^T * sqrt(dx)
    gramT<0, 0, 0, NXg>(stream, K, K, GP, 1.f, G, nullptr);
    k_chol64<<<dim3(1), dim3(64), 0, stream>>>(G);
    k_trsm<<<dim3(NXg / 256), B256, 0, stream>>>(K, K, NXg, G, isDX);
  };
  kstep(Ke, vpm_e, vmm_e, Vl_e, Vr_e, sve, VMe, epe, eme, NUEf);
  k_transcale<<<dim3(16), B256, 0, stream>>>(G, Se_c, 1, sDX);
  kstep(Ki, vpm_i, vmm_i, Vl_i, Vr_i, svi, VMi, epi, emi, NUIf);
  k_transcale<<<dim3(16), B256, 0, stream>>>(G, Si_c, 1, sDX);
  // from here on: Ke/Ki buffers hold the new Xe/Xi bases.

  // ---------- S step ----------
  k_fluxt1<<<dim3(1), dim3(64), 0, stream>>>(Ki, Si_c, t1v);
  k_dot64<<<dim3(1), dim3(64), 0, stream>>>(t1v, wvi, flx);
  k_matvec64<<<dim3(1), dim3(64), 0, stream>>>(Se_c, sve, t64, 1.f);
  k_coldot<<<dim3(NXg / 256), B256, 0, stream>>>(Ke, NXg, t64, ZEf * DVf, 0.f, rho);
  k_matvec64<<<dim3(1), dim3(64), 0, stream>>>(Si_c, svi, t64, 1.f);
  k_coldot<<<dim3(NXg / 256), B256, 0, stream>>>(Ki, NXg, t64, ZIf * DVf, 1.f, rho);
  k_scan_E<<<dim3(1), dim3(1024), 0, stream>>>(rho, Eb, mean);
  k_submean<<<dim3(NXg / 256), B256, 0, stream>>>(Eb, mean);
  k_epem<<<dim3(NXg / 256), B256, 0, stream>>>(Eb, epe, eme, OCTf * ZEf / AEf);
  k_epem<<<dim3(NXg / 256), B256, 0, stream>>>(Eb, epi, emi, OCTf * ZIf / AIf);
  auto sstep = [&](const float* X, float* S, const float* vpm, const float* vmm,
                   const float* Vl, const float* Vr, const float* sv,
                   const float* VM, const float* ep, const float* em, float nu0) {
    gemmT<0, 0, 0, NXg>(stream, S, 1, X, BUFX, 1.f, 0.f, nullptr);  // Kf = S^T X
    gramT<1, -1, 0, NXg>(stream, X, BUFX, GP, 1.f, Klm, nullptr);
    gramT<1, 0, 1, NXg>(stream, X, BUFX, GP, 1.f, Krm, nullptr);
    gramT<2, 0, 0, NXg>(stream, X, X, GP, DXf, Epm, ep);
    gramT<2, 0, 0, NXg>(stream, X, X, GP, DXf, Emm, em);
    mm(Klm, 0, vpm, 1, T1, 1.f, 0.f);
    mm(Krm, 0, vmm, 1, T1, 1.f, 1.f);
    mm(S, 0, Vl, 1, T2, 1.f, 0.f);
    mm(Epm, 0, T2, 0, T1, 1.f, 1.f);
    mm(S, 0, Vr, 1, T2, 1.f, 0.f);
    mm(Emm, 0, T2, 0, T1, 1.f, 1.f);
    k_matvec64<<<dim3(1), dim3(64), 0, stream>>>(S, sv, t64, 1.f);
    k_coldot<<<dim3(NXg / 256), B256, 0, stream>>>(X, NXg, t64, DVf, 0.f, nbuf);
    k_makew<<<dim3(NXg / 256), B256, 0, stream>>>(nbuf, gsh, flx, nu0, wvec);
    k_rowdot<<<dim3(64), B256, 0, stream>>>(X, NXg, wvec, 0, 0.f, DXf, Xng);
    gramT<0, 0, 0, NXg>(stream, X, X, GP, nu0 * DXf, Xnm, nullptr);
    mm(Xnm, 0, S, 0, T1, 1.f, 1.f);
    k_finishS<<<dim3(16), B256, 0, stream>>>(S, T1, Xng, VM);
  };
  sstep(Ke, Se_c, vpm_e, vmm_e, Vl_e, Vr_e, sve, VMe, epe, eme, NUEf);
  sstep(Ki, Si_c, vpm_i, vmm_i, Vl_i, Vr_i, svi, VMi, epi, emi, NUIf);

  // ---------- L step ----------
  gemmT<0, 0, 0, NVg>(stream, Se_c, 0, Ve, Le, 1.f, 0.f, nullptr);
  gemmT<0, 0, 0, NVg>(stream, Si_c, 0, Vi, Li, 1.f, 0.f, nullptr);
  k_fluxt1<<<dim3(1), dim3(64), 0, stream>>>(Ki, Si_c, t1v);
  k_dot64<<<dim3(1), dim3(64), 0, stream>>>(t1v, wvi, flx);
  k_rowdot<<<dim3(64), B256, 0, stream>>>(Le, NVg, nullptr, 3, 0.f, 1.f, lse);
  k_rowdot<<<dim3(64), B256, 0, stream>>>(Li, NVg, nullptr, 3, 0.f, 1.f, lsi);
  k_coldot<<<dim3(NXg / 256), B256, 0, stream>>>(Ke, NXg, lse, ZEf * DVf, 0.f, rho);
  k_coldot<<<dim3(NXg / 256), B256, 0, stream>>>(Ki, NXg, lsi, ZIf * DVf, 1.f, rho);
  k_scan_E<<<dim3(1), dim3(1024), 0, stream>>>(rho, Eb, mean);
  k_submean<<<dim3(NXg / 256), B256, 0, stream>>>(Eb, mean);
  k_epem<<<dim3(NXg / 256), B256, 0, stream>>>(Eb, epe, eme, OCTf * ZEf / AEf);
  k_epem<<<dim3(NXg / 256), B256, 0, stream>>>(Eb, epi, emi, OCTf * ZIf / AIf);
  auto lstep = [&](const float* X, float* L, const float* lsum, const float* ep,
                   const float* em, float Am, float nu0, float* Sout) {
    // internal QR of L: V = Lc^{-1} L / sqrt(dv), Sq = Lc * sqrt(dv)
    gramT<0, 0, 0, NVg>(stream, L, L, GP, 1.f, G, nullptr);
    k_chol64<<<dim3(1), dim3(64), 0, stream>>>(G);
    k_trsm<<<dim3(NVg / 256), B256, 0, stream>>>(L, Vt, NVg, G, isDV);
    k_transcale<<<dim3(16), B256, 0, stream>>>(G, Sq, 0, sDV);
    gemmT<0, 0, 0, NXg>(stream, Sq, 1, X, BUFX, 1.f, 0.f, nullptr);  // Kf
    gramT<1, -1, 0, NXg>(stream, X, BUFX, GP, 1.f, Klm, nullptr);
    gramT<1, 0, 1, NXg>(stream, X, BUFX, GP, 1.f, Krm, nullptr);
    gramT<2, 0, 0, NXg>(stream, X, X, GP, DXf, Epm, ep);
    gramT<2, 0, 0, NXg>(stream, X, X, GP, DXf, Emm, em);
    gemmT<2, 0, 0, NVg>(stream, Klm, 0, Vt, RHSL, -1.f, 0.f, vpl);
    gemmT<2, 0, 0, NVg>(stream, Krm, 0, Vt, RHSL, -1.f, 1.f, vmn);
    gemmT<1, -1, 0, NVg>(stream, Epm, 0, L, RHSL, -1.f / DVf, 1.f, nullptr);
    gemmT<1, 0, 1, NVg>(stream, Emm, 0, L, RHSL, -1.f / DVf, 1.f, nullptr);
    k_coldot<<<dim3(NXg / 256), B256, 0, stream>>>(X, NXg, lsum, DVf, 0.f, nbuf);
    k_makew<<<dim3(NXg / 256), B256, 0, stream>>>(nbuf, gsh, flx, nu0, wvec);
    k_rowdot<<<dim3(64), B256, 0, stream>>>(X, NXg, wvec, 0, 0.f, DXf, Xng);
    gramT<0, 0, 0, NXg>(stream, X, X, GP, nu0 * DXf, Xnm, nullptr);
    gemmT<0, 0, 0, NVg>(stream, Xnm, 0, L, RHSL, -1.f, 1.f, nullptr);
    k_finishL<<<dim3(NVg / 256, 64), B256, 0, stream>>>(L, RHSL, Xng, Am);
    // final QR: S = Lc * sqrt(dv), V = Lc^{-1} L / sqrt(dv) (in place)
    gramT<0, 0, 0, NVg>(stream, L, L, GP, 1.f, G, nullptr);
    k_chol64<<<dim3(1), dim3(64), 0, stream>>>(G);
    k_transcale<<<dim3(16), B256, 0, stream>>>(G, Sout, 0, sDV);
    k_trsm<<<dim3(NVg / 256), B256, 0, stream>>>(L, L, NVg, G, isDV);
  };
  lstep(Ke, Le, lse, epe, eme, AEf, NUEf, Se_c);
  lstep(Ki, Li, lsi, epi, emi, AIf, NUIf, Si_c);

  // ---------- write output: stack([cat(Xe,Se,Ve)], [cat(Xi,Si,Vi)]) ----------
  k_copyout<<<dim3(NXg / 256, 64), B256, 0, stream>>>(Ke, NXg, out, 0);
  k_copyout<<<dim3(1, 64), B256, 0, stream>>>(Se_c, 64, out, NXg);
  k_copyout<<<dim3(NVg / 256, 64), B256, 0, stream>>>(Le, NVg, out, NXg + 64);
  float* outi = out + (size_t)64 * OST;
  k_copyout<<<dim3(NXg / 256, 64), B256, 0, stream>>>(Ki, NXg, outi, 0);
  k_copyout<<<dim3(1, 64), B256, 0, stream>>>(Si_c, 64, outi, NXg);
  k_copyout<<<dim3(NVg / 256, 64), B256, 0, stream>>>(Li, NVg, outi, NXg + 64);
}